// DynamicFilter_65369402245158
// MI455X (gfx1250) — compile-verified
//
#include <hip/hip_runtime.h>
#include <stdint.h>

// Problem constants (from the reference): B=2, C=2 (cand0|cand1), D=16, H=W=192, K=3.
#define BQ 2
#define DQ 16
#define HQ 192
#define WQ 192
#define TAPS 27
#define CHN 2

#define TILE_W 64
#define TILE_H 16
#define LW (TILE_W + 2)                 // 66 (halo)
#define LH (TILE_H + 2)                 // 18 (halo)
#define LDS_N (CHN * 3 * LH * LW)       // 7128 floats = 28512 B

typedef float v4f __attribute__((ext_vector_type(4)));

__global__ __launch_bounds__(256) void dynfilter_k(
    const float* __restrict__ cand0,
    const float* __restrict__ cand1,
    const float* __restrict__ filt,
    float* __restrict__ out)
{
    __shared__ float tile[CHN][3][LH][LW];

    const int tid = threadIdx.x;
    const int w0  = blockIdx.x * TILE_W;
    const int h0  = blockIdx.y * TILE_H;
    const int bz  = blockIdx.z;          // b*DQ + d
    const int b   = bz >> 4;             // DQ == 16
    const int d   = bz & 15;

    // ---------------------------------------------------------------
    // Stage the 2-channel, 3-deep halo tile of `combined` into LDS.
    // In-bounds lanes: CDNA5 async global->LDS copy (ASYNCcnt path).
    // Out-of-bounds halo lanes: ds_store zero (implements padding=1).
    // Disjoint LDS addresses, so the two paths cannot race.
    // ---------------------------------------------------------------
    float* tflat = &tile[0][0][0][0];
    const unsigned lds_base = (unsigned)(uintptr_t)tflat;  // LDS byte offset

    for (int idx = tid; idx < LDS_N; idx += 256) {
        int x = idx % LW;
        int r = idx / LW;
        int y = r % LH;  r /= LH;
        int dz = r % 3;
        int c  = r / 3;
        int gw = w0 + x - 1;
        int gh = h0 + y - 1;
        int gd = d  + dz - 1;
        bool ok = ((unsigned)gw < WQ) & ((unsigned)gh < HQ) & ((unsigned)gd < DQ);
        if (ok) {
            const float* src = (c == 0 ? cand0 : cand1)
                             + (((size_t)b * DQ + gd) * HQ + gh) * (size_t)WQ + gw;
            unsigned      ldsaddr = lds_base + (unsigned)idx * 4u;
            unsigned long long ga = (unsigned long long)(uintptr_t)src;
            // GV mode: per-lane 64-bit global address, LDS dest offset in VGPR.
            asm volatile("global_load_async_to_lds_b32 %0, %1, off"
                         :: "v"(ldsaddr), "v"(ga)
                         : "memory");
        } else {
            tflat[idx] = 0.0f;           // ds_store_b32 (DScnt path)
        }
    }
    // Wait for this wave's async LDS writes, then workgroup barrier
    // (barrier also covers the DScnt zero-stores).
    asm volatile("s_wait_asynccnt 0x0" ::: "memory");
    __syncthreads();

    // ---------------------------------------------------------------
    // Compute: each thread owns 4 consecutive w outputs at (b,d,h,w..w+3).
    // Filters are the one-shot 255 MB stream -> b128 non-temporal loads.
    // ---------------------------------------------------------------
    const int xq = tid & 15;             // 16 quads across W
    const int yy = tid >> 4;             // 16 rows
    const int wb = xq << 2;              // 0..60, tile-local w of first output
    const int h  = h0 + yy;
    const int w  = w0 + wb;

    const size_t plane = (size_t)DQ * HQ * WQ;                 // channel stride
    const size_t fbase = ((size_t)b * (CHN * TAPS) * DQ + d) * ((size_t)HQ * WQ)
                       + (size_t)h * WQ + w;                   // ch = 0 base

    float acc0 = 0.f, acc1 = 0.f, acc2 = 0.f, acc3 = 0.f;

    for (int c = 0; c < CHN; ++c) {
        for (int dz = 0; dz < 3; ++dz) {
#pragma unroll
            for (int dy = 0; dy < 3; ++dy) {
                // 6 stencil inputs cover dx in 0..2 for 4 outputs.
                float xr[6];
#pragma unroll
                for (int k = 0; k < 6; ++k)
                    xr[k] = tile[c][dz][yy + dy][wb + k];
#pragma unroll
                for (int dx = 0; dx < 3; ++dx) {
                    const int ch = c * TAPS + dz * 9 + dy * 3 + dx;
                    const v4f f = __builtin_nontemporal_load(
                        (const v4f*)(filt + fbase + (size_t)ch * plane));
                    acc0 = fmaf(xr[dx + 0], f.x, acc0);
                    acc1 = fmaf(xr[dx + 1], f.y, acc1);
                    acc2 = fmaf(xr[dx + 2], f.z, acc2);
                    acc3 = fmaf(xr[dx + 3], f.w, acc3);
                }
            }
        }
    }

    v4f o;
    o.x = acc0; o.y = acc1; o.z = acc2; o.w = acc3;
    float* dst = out + (((size_t)b * DQ + d) * HQ + h) * (size_t)WQ + w;
    __builtin_nontemporal_store(o, (v4f*)dst);
}

extern "C" void kernel_launch(void* const* d_in, const int* in_sizes, int n_in,
                              void* d_out, int out_size, void* d_ws, size_t ws_size,
                              hipStream_t stream) {
    const float* cand0 = (const float*)d_in[0];
    const float* cand1 = (const float*)d_in[1];
    const float* filt  = (const float*)d_in[2];
    float*       out   = (float*)d_out;

    dim3 grid(WQ / TILE_W, HQ / TILE_H, BQ * DQ);   // 3 x 12 x 32 = 1152 blocks
    dim3 block(256);                                 // 8 wave32 per block
    hipLaunchKernelGGL(dynfilter_k, grid, block, 0, stream, cand0, cand1, filt, out);
}